// Quantizer_91001767068468
// MI455X (gfx1250) — compile-verified
//
#include <hip/hip_runtime.h>
#include <hip/hip_bf16.h>
#include <stdint.h>

#define HIDDEN 4096
#define TPB 256
#define EPT 16                 // elements per thread = HIDDEN / TPB
#define WAVES (TPB / 32)
#define ROWS_PER_BLOCK 8

// ---- gfx1250 async global->LDS path (guarded; falls back to direct loads) ----
#if defined(__has_builtin)
#  if __has_builtin(__builtin_amdgcn_global_load_async_to_lds_b128) && \
      __has_builtin(__builtin_amdgcn_s_wait_asynccnt)
#    define USE_ASYNC 1
#  endif
#endif
#ifndef USE_ASYNC
#  define USE_ASYNC 0
#endif

typedef int v4i __attribute__((vector_size(16)));
typedef __attribute__((address_space(1))) v4i Gv4i;   // global
typedef __attribute__((address_space(3))) v4i Lv4i;   // LDS

__device__ __forceinline__ float quant_to_f(float v, float inv) {
    float r = rintf(v * inv);               // v_rndne_f32: round half to even
    r = fminf(fmaxf(r, -8.0f), 7.0f);       // clip to int4 range
    return r;
}

__global__ __launch_bounds__(TPB) void
quantize_int4_kernel(const float* __restrict__ x,
                     float* __restrict__ out_packed,   // rows * HIDDEN/2 floats
                     float* __restrict__ out_scales,   // rows floats
                     int rows) {
    const int tid  = threadIdx.x;
    const int lane = tid & 31;
    const int wave = tid >> 5;

    __shared__ float partials[WAVES];
#if USE_ASYNC
    __shared__ float buf[2][HIDDEN];
#endif

    int row = blockIdx.x;
    if (row >= rows) return;
    const int stride = gridDim.x;

#if USE_ASYNC
    // Issue one row's worth of async loads into LDS slot `s` (4 x b128 / thread).
    auto issue = [&](int s, int r) {
        const float* g = x + (size_t)r * HIDDEN + tid * EPT;
        float* l = &buf[s][tid * EPT];
        __builtin_amdgcn_global_load_async_to_lds_b128((Gv4i*)g, (Lv4i*)l, 0, 0);
        __builtin_amdgcn_global_load_async_to_lds_b128((Gv4i*)g, (Lv4i*)l, 16, 0);
        __builtin_amdgcn_global_load_async_to_lds_b128((Gv4i*)g, (Lv4i*)l, 32, 0);
        __builtin_amdgcn_global_load_async_to_lds_b128((Gv4i*)g, (Lv4i*)l, 48, 0);
    };
    int cur = 0;
    issue(0, row);                      // prologue: prefetch first row
#endif

    for (; row < rows; row += stride) {
        int next = row + stride;
        if (next >= rows) next = row;   // clamped redundant prefetch on last iter

#if USE_ASYNC
        issue(cur ^ 1, next);                       // overlap next row's DMA
        __builtin_amdgcn_s_wait_asynccnt(4);        // current row's 4 ops done
        const float4* src = (const float4*)&buf[cur][tid * EPT];
#else
        const float4* src = (const float4*)(x + (size_t)row * HIDDEN + tid * EPT);
#endif
        float4 v0 = src[0];
        float4 v1 = src[1];
        float4 v2 = src[2];
        float4 v3 = src[3];

        // local max|x| over this thread's 16 elements
        float m = fabsf(v0.x);
        m = fmaxf(m, fabsf(v0.y)); m = fmaxf(m, fabsf(v0.z)); m = fmaxf(m, fabsf(v0.w));
        m = fmaxf(m, fabsf(v1.x)); m = fmaxf(m, fabsf(v1.y)); m = fmaxf(m, fabsf(v1.z)); m = fmaxf(m, fabsf(v1.w));
        m = fmaxf(m, fabsf(v2.x)); m = fmaxf(m, fabsf(v2.y)); m = fmaxf(m, fabsf(v2.z)); m = fmaxf(m, fabsf(v2.w));
        m = fmaxf(m, fabsf(v3.x)); m = fmaxf(m, fabsf(v3.y)); m = fmaxf(m, fabsf(v3.z)); m = fmaxf(m, fabsf(v3.w));

        // wave32 reduction
        #pragma unroll
        for (int off = 16; off >= 1; off >>= 1)
            m = fmaxf(m, __shfl_xor(m, off, 32));

        if (lane == 0) partials[wave] = m;
        __syncthreads();

        float mm = partials[0];
        #pragma unroll
        for (int w = 1; w < WAVES; ++w) mm = fmaxf(mm, partials[w]);

        const float scale = mm * (1.0f / 7.0f);     // max|x|/7 (clip_ratio = 1)
        const float inv   = 7.0f / mm;              // == 1/scale
        if (tid == 0) out_scales[row] = scale;

        float vals[EPT] = { v0.x, v0.y, v0.z, v0.w,
                            v1.x, v1.y, v1.z, v1.w,
                            v2.x, v2.y, v2.z, v2.w,
                            v3.x, v3.y, v3.z, v3.w };

        float outs[EPT / 2];
        #pragma unroll
        for (int i = 0; i < EPT / 2; ++i) {
            int q0 = (int)quant_to_f(vals[2 * i + 0], inv);  // even -> low nibble
            int q1 = (int)quant_to_f(vals[2 * i + 1], inv);  // odd  -> high nibble
            int8_t b = (int8_t)(((q1 & 0xF) << 4) | (q0 & 0xF));
            outs[i] = (float)b;                               // harness views output as f32
        }

        float4* dst = (float4*)(out_packed + (size_t)row * (HIDDEN / 2) + tid * (EPT / 2));
        dst[0] = make_float4(outs[0], outs[1], outs[2], outs[3]);
        dst[1] = make_float4(outs[4], outs[5], outs[6], outs[7]);

        __syncthreads();                // partials reused next iteration
#if USE_ASYNC
        cur ^= 1;
#endif
    }
}

extern "C" void kernel_launch(void* const* d_in, const int* in_sizes, int n_in,
                              void* d_out, int out_size, void* d_ws, size_t ws_size,
                              hipStream_t stream) {
    (void)n_in; (void)out_size; (void)d_ws; (void)ws_size;
    const float* x = (const float*)d_in[0];
    const int rows = in_sizes[0] / HIDDEN;          // 8192 for the reference shapes

    float* out_packed = (float*)d_out;               // rows * HIDDEN/2 values (int8 as f32)
    float* out_scales = out_packed + (size_t)rows * (HIDDEN / 2);

    int grid = rows / ROWS_PER_BLOCK;               // persistent blocks, 8 rows each
    if (grid < 1) grid = 1;

    quantize_int4_kernel<<<grid, TPB, 0, stream>>>(x, out_packed, out_scales, rows);
}